// FilteredNoise_19705309954209
// MI455X (gfx1250) — compile-verified
//
#include <hip/hip_runtime.h>
#include <math.h>

typedef __attribute__((ext_vector_type(2))) float v2f;
typedef __attribute__((ext_vector_type(8))) float v8f;

#define FRAME_LEN 64
#define NF 65            // rfft bins (F)
#define KPAD 68          // K padded to 17 * 4
#define NIR 129          // impulse response length (2F-1)
#define LCONV 192        // NIR + FRAME_LEN - 1
#define T_USED 1334      // frames per batch that actually reach the output
#define T_FULL 4000
#define OUT_PER_B 256000 // T_FULL * FRAME_LEN
#define HW_STRIDE 130
#define WAVES_PER_BLOCK 4
#define FRAMES_PER_WAVE 16
#define FRAMES_PER_BLOCK (WAVES_PER_BLOCK * FRAMES_PER_WAVE) // 64
#define TOTAL_FRAMES (32 * T_USED)                           // 42688

__global__ __launch_bounds__(128)
void filtered_noise_kernel(const float* __restrict__ H,
                           const float* __restrict__ noise,
                           float* __restrict__ out)
{
    // hw impulse responses (post-GEMM), per-wave private regions
    __shared__ float hwlds[WAVES_PER_BLOCK * FRAMES_PER_WAVE * HW_STRIDE]; // 33280 B
    // staging: first H rows padded to KPAD, later reused (same stride) for noise x
    __shared__ float stage[WAVES_PER_BLOCK * FRAMES_PER_WAVE * KPAD];      // 17408 B
    __shared__ float csc[NIR];   // (2/129) * cos(2*pi*p/129)
    __shared__ float winl[NIR];  // 0.5 - 0.5*cos(2*pi*n/129)

    const int tid  = threadIdx.x;
    const int wave = tid >> 5;
    const int lane = tid & 31;
    const int frameBase = blockIdx.x * FRAMES_PER_BLOCK + wave * FRAMES_PER_WAVE;

    const float w0 = 6.28318530717958647692f / 129.0f;

    // ---- per-block constant tables (only place cosf appears) ----
    for (int p = tid; p < NIR; p += 128) {
        float c = cosf(w0 * (float)p);
        csc[p]  = (2.0f / 129.0f) * c;
        winl[p] = 0.5f - 0.5f * c;
    }

    // ---- stage this wave's 16 H rows into LDS, zero-padded to K = 68 ----
    float* Hs = &stage[wave * FRAMES_PER_WAVE * KPAD];
    for (int i = lane; i < FRAMES_PER_WAVE * KPAD; i += 32) {
        int fr = i / KPAD, k = i - fr * KPAD;
        int f  = frameBase + fr;
        float v = 0.0f;
        if (k < NF && f < TOTAL_FRAMES) {
            int b = f / T_USED, t = f - b * T_USED;
            v = H[(size_t)(b * T_FULL + t) * NF + k];
        }
        Hs[i] = v;
    }

    // prefetch the noise rows this wave will convolve with later (global_prefetch_b8)
    if (lane < FRAMES_PER_WAVE) {
        int f = frameBase + lane;
        if (f < TOTAL_FRAMES) {
            int b = f / T_USED, t = f - b * T_USED;
            __builtin_prefetch(noise + (size_t)(b * T_FULL + t) * FRAME_LEN, 0, 3);
        }
    }

    __syncthreads();   // tables are block-shared

    // ---- WMMA GEMM: hw[16 x 129] = Hs[16 x 68] * C[68 x 129(pad 144)]
    // f32 A 16x4 layout: lanes 0-15 -> M=lane, K=kb+0/kb+1 ; lanes 16-31 -> K=kb+2/kb+3
    const int mrow  = lane & 15;
    const int koff  = (lane >> 4) << 1;     // 0 or 2
    const int ncol  = lane & 15;
    const int mbase = (lane >> 4) << 3;
    const char* cscB = (const char*)csc;    // byte-addressed table

    for (int nt = 0; nt < 9; ++nt) {
        const int n    = nt * 16 + ncol;
        const int nmod = (n < NIR) ? n : (n - NIR);
        const float win = winl[nmod];       // column scale, folded in at store time

        // incremental phases (byte offsets): p advances by d = (4n) mod 129 each k-step
        int q0 = ((koff       * n) % NIR) * 4;
        int q1 = (((koff + 1) * n) % NIR) * 4;
        const int dq = ((4 * n) % NIR) * 4;

        v8f acc = {};
        #pragma unroll
        for (int kk = 0; kk < 17; ++kk) {
            const int k0 = kk * 4 + koff;
            const v2f a = *(const v2f*)&Hs[mrow * KPAD + k0];   // ds_load_b64
            float b0 = *(const float*)(cscB + q0);
            if (k0 == 0) b0 *= 0.5f;        // only possible in the kk==0 copy
            const float b1 = *(const float*)(cscB + q1);
            q0 += dq; if (q0 >= NIR * 4) q0 -= NIR * 4;
            q1 += dq; if (q1 >= NIR * 4) q1 -= NIR * 4;
            v2f bf; bf.x = b0; bf.y = b1;
            acc = __builtin_amdgcn_wmma_f32_16x16x4_f32(
                      false, a, false, bf, (short)0, acc, false, false);
        }
        // C tile -> LDS (x window). Lanes 0-15 hold M=v, lanes 16-31 hold M=v+8.
        if (n < NIR) {
            #pragma unroll
            for (int v = 0; v < 8; ++v)
                hwlds[(wave * FRAMES_PER_WAVE + mbase + v) * HW_STRIDE + n] = win * acc[v];
        }
    }

    // ---- restage noise as x = 2*u - 1 into the (now free) H staging area ----
    for (int i = lane; i < FRAMES_PER_WAVE * FRAME_LEN; i += 32) {
        int fr = i >> 6, m = i & 63;
        int f  = frameBase + fr;
        float v = 0.0f;
        if (f < TOTAL_FRAMES) {
            int b = f / T_USED, t = f - b * T_USED;
            v = 2.0f * noise[(size_t)(b * T_FULL + t) * FRAME_LEN + m] - 1.0f;
        }
        Hs[fr * KPAD + m] = v;
    }

    __syncthreads();

    // ---- convolution, one frame at a time, lanes span l = j*32 + lane ----
    for (int fr = 0; fr < FRAMES_PER_WAVE; ++fr) {
        const int f = frameBase + fr;
        if (f >= TOTAL_FRAMES) break;                  // uniform per wave
        const int b = f / T_USED, t = f - b * T_USED;
        float* orow = out + (size_t)b * OUT_PER_B + (size_t)t * LCONV;
        const int jlim = OUT_PER_B - t * LCONV;        // trims frame t = 1333
        const float* xrow  = &Hs[fr * KPAD];
        const float* hwrow = &hwlds[(wave * FRAMES_PER_WAVE + fr) * HW_STRIDE];
        #pragma unroll
        for (int j = 0; j < 6; ++j) {
            const int l   = j * 32 + lane;
            const int mlo = (32 * j - 128 > 0) ? 32 * j - 128 : 0;   // compile-time
            const int mhi = (32 * j + 31 < 63) ? 32 * j + 31 : 63;   // compile-time
            float acc = 0.0f;
            for (int m = mlo; m <= mhi; ++m) {
                const int idx = l - m;
                const bool ok = (unsigned)idx <= 128u;
                const float hv = hwrow[ok ? idx : 0];
                acc = fmaf(xrow[m], ok ? hv : 0.0f, acc);
            }
            if (l < jlim) orow[l] = 0.01f * acc;       // coalesced b32 stores
        }
    }
}

extern "C" void kernel_launch(void* const* d_in, const int* in_sizes, int n_in,
                              void* d_out, int out_size, void* d_ws, size_t ws_size,
                              hipStream_t stream) {
    const float* H     = (const float*)d_in[0];   // [32, 4000, 65]
    const float* noise = (const float*)d_in[1];   // [32, 4000, 64]
    float* out = (float*)d_out;                   // [32, 256000]
    (void)in_sizes; (void)n_in; (void)out_size; (void)d_ws; (void)ws_size;

    // 42688 contributing frames, 64 per block -> exactly 667 blocks
    dim3 grid(TOTAL_FRAMES / FRAMES_PER_BLOCK), block(128);
    filtered_noise_kernel<<<grid, block, 0, stream>>>(H, noise, out);
}